// RCModel_12300786336360
// MI455X (gfx1250) — compile-verified
//
#include <hip/hip_runtime.h>

// ---------------------------------------------------------------------------
// RCModel RK4 rollout as a linear-propagator recurrence:
//   x_{n+1} = P x_n + f_n,  P = I + M + M^2/2 + M^3/6 + M^4/24,  M = h*A
//   f_n = T1*w1 + T2*w2 + T3*w3 + T4*w4 + d    (exact for linear ODE + RK4)
// Chunked parallel recurrence (64 chunks x 64 steps); all heavy work is dense
// fp32 GEMM via V_WMMA_F32_16X16X4_F32, with cooperative LDS tiling:
// 256-thread blocks = 8 waves as 2x4 tiles, double-buffered 32xKC / 64xKC
// LDS slabs (padded against bank conflicts), global->LDS prefetch one chunk
// ahead so L2 latency hides under the 16-wmma compute phase.
// ---------------------------------------------------------------------------

typedef float v2f __attribute__((ext_vector_type(2)));
typedef float v4f __attribute__((ext_vector_type(4)));
typedef float v8f __attribute__((ext_vector_type(8)));

#define NS   2048      // state dimension
#define TS   4096      // output time points
#define LCH  64        // steps per chunk
#define CCH  64        // chunks (= GEMM column count in passes 1/3)
#define DT_H 30.0f     // uniform t_eval step
#define KC   64        // GEMM K-chunk staged in LDS
#define KCP  (KC + 4)  // padded LDS row stride: banks (4*l16+k0)%64 all distinct

// ---------------------------------------------------------------------------
// GEMM: D(2048 x Ncols) = A(2048 x 2048, row-major) * B(2048 x Ncols).
// B addressed column-major: elem(k,c) = Bb[c*sC + k].
// MODE 0: Dt[c*2048+m] = alpha*acc (+I)                   (E1t / E2t)
// MODE 1: D[m*2048+c] = v  AND  D2[c*2048+m] = v          (P, squarings)
// MODE 2: out[(c*64+j+1)*NS+m] = acc + F[(c*64+j)*NS+m]   (pass 1 scatter)
// MODE 3: D[c*NS+m] = acc;  D2[(c*64+j)*NS+m] += acc      (pass 3 emit)
// Grid: 64 row-blocks x (tilesN/4) col-blocks, exact (no guards -> EXEC all 1s)
// ---------------------------------------------------------------------------
template <int MODE>
__global__ __launch_bounds__(256) void gemm_wmma(
    const float* __restrict__ A, const float* __restrict__ Bb, long sC,
    float alpha, int addI, int jstep,
    float* __restrict__ D, float* __restrict__ D2, const float* __restrict__ Fb)
{
  __shared__ float As[2][32 * KCP];
  __shared__ float Bs[2][64 * KCP];

  int tid  = (int)threadIdx.x;
  int bm   = (int)blockIdx.x & 63;       // 64 row-blocks (128 tiles / 2)
  int bn   = (int)blockIdx.x >> 6;
  int wid  = tid >> 5;
  int wtm  = wid >> 2;                   // 0..1  (tile row within block)
  int wtn  = wid & 3;                    // 0..3  (tile col within block)
  int lane = tid & 31;
  int half = lane >> 4;                  // 0: K=0,1   1: K=2,3
  int l16  = lane & 15;

  int  rBase = bm * 32;                  // output rows [rBase, rBase+32)
  long cBase = (long)bn * 64;            // output cols [cBase, cBase+64)

  // staging thread roles (A: 512 float4, B: 1024 float4 per chunk)
  v4f pa[2], pb[4];
  #pragma unroll
  for (int i = 0; i < 2; ++i) {          // chunk 0 -> regs
    int idx = tid + 256 * i; int r = idx >> 4; int f = (idx & 15) << 2;
    pa[i] = *(const v4f*)(A + (long)(rBase + r) * NS + f);
  }
  #pragma unroll
  for (int i = 0; i < 4; ++i) {
    int idx = tid + 256 * i; int c = idx >> 4; int f = (idx & 15) << 2;
    pb[i] = *(const v4f*)(Bb + (cBase + c) * sC + f);
  }
  #pragma unroll
  for (int i = 0; i < 2; ++i) {          // regs -> LDS buf 0
    int idx = tid + 256 * i; int r = idx >> 4; int f = (idx & 15) << 2;
    *(v4f*)&As[0][r * KCP + f] = pa[i];
  }
  #pragma unroll
  for (int i = 0; i < 4; ++i) {
    int idx = tid + 256 * i; int c = idx >> 4; int f = (idx & 15) << 2;
    *(v4f*)&Bs[0][c * KCP + f] = pb[i];
  }
  __syncthreads();

  int aoffL = (wtm * 16 + l16) * KCP + (half << 1);
  int boffL = (wtn * 16 + l16) * KCP + (half << 1);

  v8f acc = {0.f, 0.f, 0.f, 0.f, 0.f, 0.f, 0.f, 0.f};
  const int NCHUNK = NS / KC;            // 32
  for (int ch = 0; ch < NCHUNK; ++ch) {
    int cur = ch & 1;
    if (ch + 1 < NCHUNK) {               // prefetch next chunk into regs
      int kc = (ch + 1) * KC;
      #pragma unroll
      for (int i = 0; i < 2; ++i) {
        int idx = tid + 256 * i; int r = idx >> 4; int f = (idx & 15) << 2;
        pa[i] = *(const v4f*)(A + (long)(rBase + r) * NS + kc + f);
      }
      #pragma unroll
      for (int i = 0; i < 4; ++i) {
        int idx = tid + 256 * i; int c = idx >> 4; int f = (idx & 15) << 2;
        pb[i] = *(const v4f*)(Bb + (cBase + c) * sC + kc + f);
      }
    }
    const float* ap = &As[cur][aoffL];
    const float* bp = &Bs[cur][boffL];
    #pragma unroll
    for (int k = 0; k < KC; k += 4) {    // 16 wmma from LDS, imm offsets
      v2f a = *(const v2f*)(ap + k);
      v2f b = *(const v2f*)(bp + k);
      acc = __builtin_amdgcn_wmma_f32_16x16x4_f32(
          false, a, false, b, (short)0, acc, false, false);
    }
    if (ch + 1 < NCHUNK) {               // rotate: store nxt, one barrier
      int nxt = cur ^ 1;
      #pragma unroll
      for (int i = 0; i < 2; ++i) {
        int idx = tid + 256 * i; int r = idx >> 4; int f = (idx & 15) << 2;
        *(v4f*)&As[nxt][r * KCP + f] = pa[i];
      }
      #pragma unroll
      for (int i = 0; i < 4; ++i) {
        int idx = tid + 256 * i; int c = idx >> 4; int f = (idx & 15) << 2;
        *(v4f*)&Bs[nxt][c * KCP + f] = pb[i];
      }
      __syncthreads();
    }
  }

  int tm = bm * 2 + wtm;
  int tn = (int)bn * 4 + wtn;
  int bcol  = tn * 16 + l16;             // global output column
  int mBase = tm * 16 + half * 8;        // C/D vgpr r -> row mBase+r
  if (MODE == 0) {
    #pragma unroll
    for (int r = 0; r < 8; ++r) {
      int m = mBase + r;
      float v = alpha * acc[r];
      if (addI && m == bcol) v += 1.0f;
      D[(long)bcol * NS + m] = v;        // col-major, contiguous
    }
  } else if (MODE == 1) {
    #pragma unroll
    for (int r = 0; r < 8; ++r) {
      int m = mBase + r;
      float v = alpha * acc[r];
      if (addI && m == bcol) v += 1.0f;
      D [(long)m * NS + bcol] = v;       // row-major (A-operand use)
      D2[(long)bcol * NS + m] = v;       // col-major (B-operand use)
    }
  } else if (MODE == 2) {
    long nrow = (long)bcol * LCH + jstep;
    if (nrow + 1 < TS) {                 // skip unused y_{63,64}
      #pragma unroll
      for (int r = 0; r < 8; ++r) {
        int m = mBase + r;
        D[(nrow + 1) * NS + m] = acc[r] + Fb[nrow * NS + m];
      }
    }
  } else { // MODE == 3
    long n = (long)bcol * LCH + jstep;
    #pragma unroll
    for (int r = 0; r < 8; ++r) {
      int m = mBase + r;
      float v = acc[r];
      D [(long)bcol * NS + m] = v;       // U_j (col-major)
      D2[n * NS + m] += v;               // x_n = Y_j + P^j z_c
    }
  }
}

// ---------------------------------------------------------------------------
// Tiled transpose with scale + identity: Out[c*NS + r] = alpha*In[r*NS + c] + I
// ---------------------------------------------------------------------------
__global__ __launch_bounds__(256) void transpose_si_kernel(
    const float* __restrict__ In, float* __restrict__ Out, float alpha, int addI)
{
  __shared__ float tile[32][33];
  int bx = (int)blockIdx.x & 63, by = (int)blockIdx.x >> 6;
  int tx = (int)(threadIdx.x & 31), ty = (int)(threadIdx.x >> 5);
  int r0 = by * 32, c0 = bx * 32;
  for (int i = ty; i < 32; i += 8)
    tile[i][tx] = In[(long)(r0 + i) * NS + c0 + tx];
  __syncthreads();
  for (int i = ty; i < 32; i += 8) {
    float v = alpha * tile[tx][i];
    if (addI && (r0 + tx) == (c0 + i)) v += 1.0f;
    Out[(long)(c0 + i) * NS + r0 + tx] = v;
  }
}

// ---------------------------------------------------------------------------
// Generic matvec: y[row] = sum_k Mat[row*rowStride + k]*x[k] (+ addv[row])
// ---------------------------------------------------------------------------
__global__ __launch_bounds__(256) void matvec_kernel(
    const float* __restrict__ Mat, long rowStride, int K,
    const float* __restrict__ x, const float* __restrict__ addv,
    float* __restrict__ y)
{
  __shared__ float red[256];
  int sub = (int)(threadIdx.x >> 6);
  int t   = (int)(threadIdx.x & 63);
  int row = (int)blockIdx.x * 4 + sub;
  const float* Mr = Mat + (long)row * rowStride;
  float s = 0.f;
  for (int k = t; k < K; k += 64) s += Mr[k] * x[k];
  red[threadIdx.x] = s;
  __syncthreads();
  for (int off = 32; off > 0; off >>= 1) {
    if (t < off) red[threadIdx.x] += red[threadIdx.x + off];
    __syncthreads();
  }
  if (t == 0) {
    float v = red[sub * 64];
    if (addv) v += addv[row];
    y[row] = v;
  }
}

// ---------- small elementwise / setup kernels ----------
__global__ void scale_kernel(const float* __restrict__ A, float* __restrict__ M,
                             float s, long n) {
  long i = (long)blockIdx.x * blockDim.x + threadIdx.x;
  if (i < n) M[i] = s * A[i];
}

__global__ void b0_extract_kernel(const float* __restrict__ Bmat,
                                  float* __restrict__ b0) {
  int i = (int)(blockIdx.x * blockDim.x + threadIdx.x);
  if (i < NS) b0[i] = Bmat[(long)i * (NS - 1)];   // B[:,0], row stride 2047
}

__global__ void combine_kernel(const float* __restrict__ b0,
                               const float* __restrict__ m1b,
                               const float* __restrict__ m2b,
                               const float* __restrict__ m3b,
                               const float* __restrict__ cv,
                               const float* __restrict__ m1c,
                               const float* __restrict__ m2c,
                               const float* __restrict__ m3c,
                               float* __restrict__ w1, float* __restrict__ w2,
                               float* __restrict__ w3, float* __restrict__ w4,
                               float* __restrict__ dv) {
  int i = (int)(blockIdx.x * blockDim.x + threadIdx.x);
  if (i < NS) {
    const float h = DT_H, h8 = h * 0.125f;
    w1[i] = h8 * (b0[i] + m1b[i] + (m2b[i] + m3b[i]) * (1.f / 3.f));
    w2[i] = h8 * (3.f * b0[i] + 2.f * m1b[i] + m2b[i]);
    w3[i] = h8 * (3.f * b0[i] + m1b[i]);
    w4[i] = h8 * b0[i];
    dv[i] = h * (cv[i] + 0.5f * m1c[i] + (1.f / 6.f) * m2c[i] +
                 (1.f / 24.f) * m3c[i]);
  }
}

// Tout at t = n*30 + q*10 (q=0..3), jnp.interp semantics (clamped, bsearch)
__global__ void tout_kernel(const float* __restrict__ tt,
                            const float* __restrict__ tv, int Kt,
                            float* __restrict__ Tq) {
  int idx = (int)(blockIdx.x * blockDim.x + threadIdx.x);
  if (idx >= TS * 4) return;
  int n = idx >> 2, q = idx & 3;
  float t = (float)n * DT_H + (float)q * (DT_H / 3.0f);
  float val;
  if (t <= tt[0]) val = tv[0];
  else if (t >= tt[Kt - 1]) val = tv[Kt - 1];
  else {
    int lo = 0, hi = Kt - 1;
    while (hi - lo > 1) {
      int mid = (lo + hi) >> 1;
      if (tt[mid] <= t) lo = mid; else hi = mid;
    }
    float t0 = tt[lo], t1 = tt[lo + 1];
    val = tv[lo] + (tv[lo + 1] - tv[lo]) * ((t - t0) / (t1 - t0));
  }
  Tq[idx] = val;
}

__global__ void forcing_kernel(const float* __restrict__ Tq,
                               const float* __restrict__ w1,
                               const float* __restrict__ w2,
                               const float* __restrict__ w3,
                               const float* __restrict__ w4,
                               const float* __restrict__ dv,
                               float* __restrict__ F) {
  long idx = (long)blockIdx.x * blockDim.x + threadIdx.x;
  if (idx < (long)TS * NS) {
    int n = (int)(idx >> 11), i = (int)(idx & (NS - 1));
    float v = 0.f;
    if (n < TS - 1)
      v = Tq[n * 4 + 0] * w1[i] + Tq[n * 4 + 1] * w2[i] +
          Tq[n * 4 + 2] * w3[i] + Tq[n * 4 + 3] * w4[i] + dv[i];
    F[idx] = v;                                  // padded step 4095 -> 0
  }
}

// Y_1 = F_0 : out[(c*64+1)] = F[(c*64)]
__global__ void y1_emit_kernel(const float* __restrict__ F,
                               float* __restrict__ out) {
  long idx = (long)blockIdx.x * blockDim.x + threadIdx.x;
  if (idx < (long)CCH * NS) {
    int c = (int)(idx >> 11), i = (int)(idx & (NS - 1));
    out[((long)c * LCH + 1) * NS + i] = F[((long)c * LCH) * NS + i];
  }
}

__global__ void z0_set_kernel(const float* __restrict__ x0,
                              float* __restrict__ Z) {
  int i = (int)(blockIdx.x * blockDim.x + threadIdx.x);
  if (i < NS) Z[i] = x0[i];
}

// x_{c*64} = z_c
__global__ void emit_j0_kernel(const float* __restrict__ Z,
                               float* __restrict__ out) {
  long idx = (long)blockIdx.x * blockDim.x + threadIdx.x;
  if (idx < (long)CCH * NS) {
    int c = (int)(idx >> 11), i = (int)(idx & (NS - 1));
    out[((long)c * LCH) * NS + i] = Z[(long)c * NS + i];
  }
}

// ---------------------------------------------------------------------------
extern "C" void kernel_launch(void* const* d_in, const int* in_sizes, int n_in,
                              void* d_out, int out_size, void* d_ws, size_t ws_size,
                              hipStream_t stream) {
  (void)n_in; (void)out_size; (void)ws_size;
  // inputs: t_eval, A, B, x0, Q_gain, Tout_times, Tout_vals (all fp32)
  const float* Ain  = (const float*)d_in[1];
  const float* Bmat = (const float*)d_in[2];
  const float* x0   = (const float*)d_in[3];
  const float* Qg   = (const float*)d_in[4];
  const float* Tt   = (const float*)d_in[5];
  const float* Tv   = (const float*)d_in[6];
  float* out = (float*)d_out;

  const long NN = (long)NS * NS;
  float* ws = (float*)d_ws;            // needs ~120 MB of scratch
  float* B0 = ws;                      // M (row), later squaring row ping
  float* B1 = B0 + NN;                 // E0t/E2t, later squaring col ping
  float* B2 = B1 + NN;                 // E1t, later squaring row pong (-> P^64)
  float* B3 = B2 + NN;                 // P (row-major)
  float* B4 = B3 + NN;                 // Pt, later squaring col pong
  float* F  = B4 + NN;                 // forcing, TS x NS
  float* Z  = F + (long)TS * NS;       // chunk-start states, [c][i]
  float* U0 = Z + (long)CCH * NS;
  float* U1 = U0 + (long)CCH * NS;
  float* vb = U1 + (long)CCH * NS;
  float* b0v = vb;        float* cv  = vb + NS;
  float* v1b = vb + 2*NS; float* v2b = vb + 3*NS; float* v3b = vb + 4*NS;
  float* v1c = vb + 5*NS; float* v2c = vb + 6*NS; float* v3c = vb + 7*NS;
  float* w1  = vb + 8*NS; float* w2  = vb + 9*NS; float* w3 = vb + 10*NS;
  float* w4  = vb + 11*NS; float* dv = vb + 12*NS;
  float* Tq  = vb + 13*NS;             // TS*4 floats

  const int TPB = 256;
  dim3 blk(TPB);
  dim3 grdNN((unsigned)((NN + TPB - 1) / TPB));
  dim3 grdGemmBig(64 * 32);            // 64 row-blocks x (128/4) col-blocks
  dim3 grdGemmCol(64);                 // 64 row-blocks x 1 col-block
  dim3 grdTr(64 * 64);
  dim3 grdMV(NS / 4);
  dim3 grdVec((NS + TPB - 1) / TPB);
  dim3 grdCN((unsigned)(((long)CCH * NS + TPB - 1) / TPB));
  dim3 grdF((unsigned)(((long)TS * NS + TPB - 1) / TPB));

  // ---- Stage A: M = h*A; P = I + M + M^2/2 + M^3/6 + M^4/24 (Horner) ----
  scale_kernel<<<grdNN, blk, 0, stream>>>(Ain, B0, DT_H, NN);
  transpose_si_kernel<<<grdTr, blk, 0, stream>>>(B0, B1, 0.25f, 1);    // E0t
  gemm_wmma<0><<<grdGemmBig, blk, 0, stream>>>(B0, B1, (long)NS,
      1.f / 3.f, 1, 0, B2, nullptr, nullptr);                          // E1t
  gemm_wmma<0><<<grdGemmBig, blk, 0, stream>>>(B0, B2, (long)NS,
      0.5f, 1, 0, B1, nullptr, nullptr);                               // E2t
  gemm_wmma<1><<<grdGemmBig, blk, 0, stream>>>(B0, B1, (long)NS,
      1.0f, 1, 0, B3, B4, nullptr);                                    // P, Pt

  // ---- Stage B: forcing basis vectors (need M=B0 alive) ----
  b0_extract_kernel<<<grdVec, blk, 0, stream>>>(Bmat, b0v);
  matvec_kernel<<<grdMV, blk, 0, stream>>>(Bmat + 1, (long)(NS - 1), NS - 2,
                                           Qg, nullptr, cv);           // c
  matvec_kernel<<<grdMV, blk, 0, stream>>>(B0, (long)NS, NS, b0v, nullptr, v1b);
  matvec_kernel<<<grdMV, blk, 0, stream>>>(B0, (long)NS, NS, v1b, nullptr, v2b);
  matvec_kernel<<<grdMV, blk, 0, stream>>>(B0, (long)NS, NS, v2b, nullptr, v3b);
  matvec_kernel<<<grdMV, blk, 0, stream>>>(B0, (long)NS, NS, cv,  nullptr, v1c);
  matvec_kernel<<<grdMV, blk, 0, stream>>>(B0, (long)NS, NS, v1c, nullptr, v2c);
  matvec_kernel<<<grdMV, blk, 0, stream>>>(B0, (long)NS, NS, v2c, nullptr, v3c);
  combine_kernel<<<grdVec, blk, 0, stream>>>(b0v, v1b, v2b, v3b, cv, v1c, v2c,
                                             v3c, w1, w2, w3, w4, dv);

  // ---- Stage C: P^64 by repeated squaring (row+col layouts each time) ----
  gemm_wmma<1><<<grdGemmBig, blk, 0, stream>>>(B3, B4, (long)NS,
      1.f, 0, 0, B0, B1, nullptr);                                     // P^2
  gemm_wmma<1><<<grdGemmBig, blk, 0, stream>>>(B0, B1, (long)NS,
      1.f, 0, 0, B2, B4, nullptr);                                     // P^4
  gemm_wmma<1><<<grdGemmBig, blk, 0, stream>>>(B2, B4, (long)NS,
      1.f, 0, 0, B0, B1, nullptr);                                     // P^8
  gemm_wmma<1><<<grdGemmBig, blk, 0, stream>>>(B0, B1, (long)NS,
      1.f, 0, 0, B2, B4, nullptr);                                     // P^16
  gemm_wmma<1><<<grdGemmBig, blk, 0, stream>>>(B2, B4, (long)NS,
      1.f, 0, 0, B0, B1, nullptr);                                     // P^32
  gemm_wmma<1><<<grdGemmBig, blk, 0, stream>>>(B0, B1, (long)NS,
      1.f, 0, 0, B2, B4, nullptr);                                     // P^64
  float* P  = B3;                      // row-major P
  float* PL = B2;                      // row-major P^64

  // ---- Stage D: forcing table F[n][i] ----
  tout_kernel<<<dim3((TS * 4 + TPB - 1) / TPB), blk, 0, stream>>>(Tt, Tv,
      in_sizes[5], Tq);
  forcing_kernel<<<grdF, blk, 0, stream>>>(Tq, w1, w2, w3, w4, dv, F);

  // ---- Pass 1: local solutions Y_j, scattered into out rows c*64+j ----
  y1_emit_kernel<<<grdCN, blk, 0, stream>>>(F, out);                   // Y_1
  for (int j = 1; j < LCH; ++j) {      // Y_{j+1} = P*Y_j + F_j (N x N x 64)
    gemm_wmma<2><<<grdGemmCol, blk, 0, stream>>>(P,
        out + (long)j * NS, (long)LCH * NS,      // elem(k,c)=out[(c*64+j)*NS+k]
        1.f, 0, j, out, nullptr, F);
  }

  // ---- Pass 2: chunk boundaries z_{c+1} = P^64 z_c + y_{c,64} ----
  z0_set_kernel<<<grdVec, blk, 0, stream>>>(x0, Z);
  for (int c = 0; c < CCH - 1; ++c) {
    matvec_kernel<<<grdMV, blk, 0, stream>>>(PL, (long)NS, NS,
        Z + (long)c * NS,
        out + (long)(c + 1) * LCH * NS,          // Y_64 column c lives here
        Z + (long)(c + 1) * NS);
  }

  // ---- Pass 3: x_{c*64+j} = Y_j + P^j z_c ----
  emit_j0_kernel<<<grdCN, blk, 0, stream>>>(Z, out);                   // j = 0
  const float* Bsrc = Z;               // U_0 = Z, col-major [c][k]
  for (int j = 1; j < LCH; ++j) {
    float* Udst = (j & 1) ? U0 : U1;
    gemm_wmma<3><<<grdGemmCol, blk, 0, stream>>>(P,
        Bsrc, (long)NS, 1.f, 0, j, Udst, out, nullptr);
    Bsrc = Udst;
  }
}